// MultiBoxLoss_5669356832492
// MI455X (gfx1250) — compile-verified
//
#include <hip/hip_runtime.h>
#include <hip/hip_bf16.h>

#define NB   128     // batch
#define NP   34125   // priors
#define NO   16      // objects per image
#define NC   21      // classes
#define NEGPOS 3
#define VAR0 0.1f
#define VAR1 0.2f
#define THRESH 0.35f

#define SPLIT 16                         // chunks per image row for occupancy
#define CHUNK ((NP + SPLIT - 1) / SPLIT) // 2133 priors per chunk

typedef float v2f __attribute__((ext_vector_type(2)));
typedef float v8f __attribute__((ext_vector_type(8)));

// ---------------------------------------------------------------------------
// Kernel 1: per-image, per-truth argmax over priors of jaccard overlap.
// One block per image, 256 threads stride over priors (priors are L2-resident).
// ---------------------------------------------------------------------------
__global__ __launch_bounds__(256) void k_best_prior(
    const float* __restrict__ targets,   // [NB, NO, 5]
    const float* __restrict__ priors,    // [NP, 4] center-size
    int* __restrict__ best_prior_idx)    // [NB, NO]
{
    const int b = blockIdx.x, tid = threadIdx.x;
    __shared__ float tx1[NO], ty1[NO], tx2[NO], ty2[NO], tarea[NO];
    __shared__ float sval[256];
    __shared__ int   sidx[256];

    if (tid < NO) {
        const float* t = targets + ((size_t)b * NO + tid) * 5;
        tx1[tid] = t[0]; ty1[tid] = t[1]; tx2[tid] = t[2]; ty2[tid] = t[3];
        tarea[tid] = (t[2] - t[0]) * (t[3] - t[1]);
    }
    __syncthreads();

    float bv[NO]; int bi[NO];
#pragma unroll
    for (int o = 0; o < NO; ++o) { bv[o] = -1.0f; bi[o] = NP; }

    for (int p = tid; p < NP; p += 256) {
        const float* pr = priors + (size_t)p * 4;
        const float cx = pr[0], cy = pr[1], w = pr[2], h = pr[3];
        const float px1 = cx - 0.5f * w, py1 = cy - 0.5f * h;
        const float px2 = cx + 0.5f * w, py2 = cy + 0.5f * h;
        const float parea = w * h;
#pragma unroll
        for (int o = 0; o < NO; ++o) {
            float iw = fminf(tx2[o], px2) - fmaxf(tx1[o], px1);
            float ih = fminf(ty2[o], py2) - fmaxf(ty1[o], py1);
            iw = fmaxf(iw, 0.0f); ih = fmaxf(ih, 0.0f);
            const float inter = iw * ih;
            const float ov = inter / (tarea[o] + parea - inter);
            if (ov > bv[o]) { bv[o] = ov; bi[o] = p; }   // strict > keeps first max
        }
    }

    // per-truth block argmax (max value, ties -> smaller prior index)
    for (int o = 0; o < NO; ++o) {
        sval[tid] = bv[o]; sidx[tid] = bi[o];
        __syncthreads();
        for (int s = 128; s > 0; s >>= 1) {
            if (tid < s) {
                const float v1 = sval[tid + s]; const int i1 = sidx[tid + s];
                if (v1 > sval[tid] || (v1 == sval[tid] && i1 < sidx[tid])) {
                    sval[tid] = v1; sidx[tid] = i1;
                }
            }
            __syncthreads();
        }
        if (tid == 0) best_prior_idx[b * NO + o] = sidx[0];
        __syncthreads();
    }
}

// ---------------------------------------------------------------------------
// Kernel 2: fused matching + encode + smooth-L1 + logsumexp-CE.
// Grid = NB * SPLIT blocks (2048) so the 437 MB stream can saturate HBM.
// Each block handles one prior-chunk of one image and writes per-chunk
// partials (deterministic; no float atomics).
// ---------------------------------------------------------------------------
__global__ __launch_bounds__(256) void k_match_losses(
    const float* __restrict__ loc_data,  // [NB, NP, 4]
    const float* __restrict__ conf_data, // [NB, NP, NC]
    const float* __restrict__ priors,    // [NP, 4]
    const float* __restrict__ targets,   // [NB, NO, 5]
    const int*   __restrict__ best_prior_idx, // [NB, NO]
    float* __restrict__ mined,           // [NB, NP]
    float* __restrict__ part_loss_l,     // [NB * SPLIT]
    float* __restrict__ part_ce_pos,     // [NB * SPLIT]
    int*   __restrict__ part_npos)       // [NB * SPLIT]
{
    const int b     = blockIdx.x / SPLIT;
    const int chunk = blockIdx.x % SPLIT;
    const int tid   = threadIdx.x;
    const int p_lo  = chunk * CHUNK;
    const int p_hi  = (p_lo + CHUNK < NP) ? (p_lo + CHUNK) : NP;

    __shared__ float tx1[NO], ty1[NO], tx2[NO], ty2[NO], tarea[NO], tlab[NO];
    __shared__ int   bpi[NO];
    __shared__ float r1[256], r2[256];
    __shared__ int   r3[256];

    if (tid < NO) {
        const float* t = targets + ((size_t)b * NO + tid) * 5;
        tx1[tid] = t[0]; ty1[tid] = t[1]; tx2[tid] = t[2]; ty2[tid] = t[3];
        tarea[tid] = (t[2] - t[0]) * (t[3] - t[1]);
        tlab[tid] = t[4];
        bpi[tid] = best_prior_idx[b * NO + tid];
    }
    __syncthreads();

    float acc_l = 0.0f, acc_cpos = 0.0f;
    int npos = 0;

    for (int p = p_lo + tid; p < p_hi; p += 256) {
        const float* pr = priors + (size_t)p * 4;
        const float cx = pr[0], cy = pr[1], w = pr[2], h = pr[3];
        const float px1 = cx - 0.5f * w, py1 = cy - 0.5f * h;
        const float px2 = cx + 0.5f * w, py2 = cy + 0.5f * h;
        const float parea = w * h;

        // best truth for this prior (max over O, ties -> smaller o)
        float bto = -1.0f; int bti = 0;
#pragma unroll
        for (int o = 0; o < NO; ++o) {
            float iw = fminf(tx2[o], px2) - fmaxf(tx1[o], px1);
            float ih = fminf(ty2[o], py2) - fmaxf(ty1[o], py1);
            iw = fmaxf(iw, 0.0f); ih = fmaxf(ih, 0.0f);
            const float inter = iw * ih;
            const float ov = inter / (tarea[o] + parea - inter);
            if (ov > bto) { bto = ov; bti = o; }
        }
        // forcing: best prior of each truth gets overlap 2.0 (last write wins)
        int forced = -1;
#pragma unroll
        for (int o = 0; o < NO; ++o) if (bpi[o] == p) forced = o;
        if (forced >= 0) { bto = 2.0f; bti = forced; }

        const int conf = (bto < THRESH) ? 0 : (int)tlab[bti];
        const bool pos = conf > 0;

        // ---- CE via register-resident logsumexp over 21 classes ----
        const float* cd = conf_data + ((size_t)b * NP + p) * NC;
        __builtin_prefetch(cd + 256 * NC, 0, 0);   // next stride (speculative)
        float xv[NC];
        float m = -3.402823466e+38f;
#pragma unroll
        for (int c = 0; c < NC; ++c) { xv[c] = cd[c]; m = fmaxf(m, xv[c]); }
        float se = 0.0f;
#pragma unroll
        for (int c = 0; c < NC; ++c) se += __expf(xv[c] - m);
        const float lse = m + __logf(se);
        const float ce  = lse - xv[conf];

        float mval;
        if (pos) {
            ++npos;
            acc_cpos += ce;
            mval = 0.0f;
            // ---- encode + smooth-L1 on the 4 coords ----
            const float mx1 = tx1[bti], my1 = ty1[bti], mx2 = tx2[bti], my2 = ty2[bti];
            const float lt0 = (0.5f * (mx1 + mx2) - cx) / (VAR0 * w);
            const float lt1 = (0.5f * (my1 + my2) - cy) / (VAR0 * h);
            const float lt2 = __logf((mx2 - mx1) / w) / VAR1;
            const float lt3 = __logf((my2 - my1) / h) / VAR1;
            const float* ld = loc_data + ((size_t)b * NP + p) * 4;
            const float lt[4] = { lt0, lt1, lt2, lt3 };
#pragma unroll
            for (int c = 0; c < 4; ++c) {
                const float d  = ld[c] - lt[c];
                const float ad = fabsf(d);
                acc_l += (ad < 1.0f) ? 0.5f * d * d : ad - 0.5f;
            }
        } else {
            mval = ce;
        }
        mined[(size_t)b * NP + p] = mval;
    }

    // block reduce (loss_l, ce_pos, npos)
    r1[tid] = acc_l; r2[tid] = acc_cpos; r3[tid] = npos;
    __syncthreads();
    for (int s = 128; s > 0; s >>= 1) {
        if (tid < s) { r1[tid] += r1[tid + s]; r2[tid] += r2[tid + s]; r3[tid] += r3[tid + s]; }
        __syncthreads();
    }
    if (tid == 0) {
        part_loss_l[blockIdx.x] = r1[0];
        part_ce_pos[blockIdx.x] = r2[0];
        part_npos[blockIdx.x]   = r3[0];
    }
}

// ---------------------------------------------------------------------------
// Kernel 2b: fold the SPLIT per-chunk partials of each row, in fixed order
// (deterministic across replays). One thread per row.
// ---------------------------------------------------------------------------
__global__ __launch_bounds__(128) void k_row_reduce(
    const float* __restrict__ part_loss_l,
    const float* __restrict__ part_ce_pos,
    const int*   __restrict__ part_npos,
    float* __restrict__ row_loss_l,
    float* __restrict__ row_ce_pos,
    int*   __restrict__ row_npos)
{
    const int b = threadIdx.x;   // one block of NB threads
    float sl = 0.0f, sc = 0.0f; int np = 0;
#pragma unroll
    for (int c = 0; c < SPLIT; ++c) {
        sl += part_loss_l[b * SPLIT + c];
        sc += part_ce_pos[b * SPLIT + c];
        np += part_npos[b * SPLIT + c];
    }
    row_loss_l[b] = sl; row_ce_pos[b] = sc; row_npos[b] = np;
}

// ---------------------------------------------------------------------------
// Kernel 3: per-row exact top-k SUM via 4-pass radix select on IEEE bits
// (all mined >= 0, so uint bit order == float order). Tie-exact:
//   topk = sum_{x>T} x + (k - count_{x>T}) * T
// Row data (136 KB) stays L2-resident across the 5 passes.
// ---------------------------------------------------------------------------
__global__ __launch_bounds__(256) void k_topk_sum(
    const float* __restrict__ mined,   // [NB, NP]
    const int*   __restrict__ row_npos,
    float* __restrict__ row_topk)
{
    const int b = blockIdx.x, tid = threadIdx.x;
    long long k = (long long)NEGPOS * row_npos[b];
    if (k > NP - 1) k = NP - 1;
    if (k <= 0) { if (tid == 0) row_topk[b] = 0.0f; return; }

    __shared__ unsigned hist[256];
    __shared__ unsigned s_prefix, s_need;
    __shared__ float    rsum[256];
    __shared__ unsigned rcnt[256];

    const float* row = mined + (size_t)b * NP;
    if (tid == 0) { s_prefix = 0u; s_need = (unsigned)k; }
    __syncthreads();

    for (int pass = 0; pass < 4; ++pass) {
        hist[tid] = 0u;
        __syncthreads();
        const int shift = 24 - 8 * pass;
        const unsigned mask_known = (pass == 0) ? 0u : (0xFFFFFFFFu << (shift + 8));
        const unsigned prefix = s_prefix;
        for (int p = tid; p < NP; p += 256) {
            const unsigned bits = __float_as_uint(row[p]);
            if ((bits & mask_known) == prefix)
                atomicAdd(&hist[(bits >> shift) & 0xFFu], 1u);
        }
        __syncthreads();
        if (tid == 0) {
            unsigned need = s_need, cum = 0u; int sel = 0;
            for (int bin = 255; bin >= 0; --bin) {
                const unsigned c2 = hist[bin];
                if (cum + c2 >= need) { sel = bin; s_need = need - cum; break; }
                cum += c2;
            }
            s_prefix = prefix | ((unsigned)sel << shift);
        }
        __syncthreads();
    }

    const unsigned tbits = s_prefix;          // bits of k-th largest value T
    const float T = __uint_as_float(tbits);
    float sg = 0.0f; unsigned cg = 0u;
    for (int p = tid; p < NP; p += 256) {
        const float v = row[p];
        if (__float_as_uint(v) > tbits) { sg += v; ++cg; }
    }
    rsum[tid] = sg; rcnt[tid] = cg;
    __syncthreads();
    for (int s = 128; s > 0; s >>= 1) {
        if (tid < s) { rsum[tid] += rsum[tid + s]; rcnt[tid] += rcnt[tid + s]; }
        __syncthreads();
    }
    if (tid == 0)
        row_topk[b] = rsum[0] + ((float)k - (float)rcnt[0]) * T;
}

// ---------------------------------------------------------------------------
// Kernel 4: final 128 -> 1 reductions on one wave using
// V_WMMA_F32_16X16X4_F32 with a ones B-matrix (exact row-sum reduction:
// products x1.0 exact, f32 accumulate). One xor-16 shuffle finishes.
// ---------------------------------------------------------------------------
__device__ __forceinline__ float wave32_sum_wmma(float s) {
    v2f a;    a[0] = s;    a[1] = 0.0f;
    v2f ones; ones[0] = 1.0f; ones[1] = 1.0f;
    v8f c = {};
    c = __builtin_amdgcn_wmma_f32_16x16x4_f32(
            false, a, false, ones, (short)0, c, false, false);
    float t = c[0] + c[1] + c[2] + c[3] + c[4] + c[5] + c[6] + c[7];
    t += __shfl_xor(t, 16, 32);
    return t;   // total across all 32 lanes, in every lane
}

__global__ __launch_bounds__(32) void k_finalize(
    const float* __restrict__ row_loss_l,
    const float* __restrict__ row_ce_pos,
    const float* __restrict__ row_topk,
    const int*   __restrict__ row_npos,
    float* __restrict__ out)
{
    const int l = threadIdx.x;
    float sl = 0.0f, sc = 0.0f, sn = 0.0f;
    for (int i = l; i < NB; i += 32) {      // uniform trip count -> EXEC stays all-ones
        sl += row_loss_l[i];
        sc += row_ce_pos[i] + row_topk[i];
        sn += (float)row_npos[i];
    }
    const float tot_l = wave32_sum_wmma(sl);
    const float tot_c = wave32_sum_wmma(sc);
    const float tot_n = wave32_sum_wmma(sn);
    if (l == 0) {
        const float N = (tot_n > 0.0f) ? tot_n : (float)NB;
        out[0] = tot_l / N;
        out[1] = tot_c / N;
    }
}

// ---------------------------------------------------------------------------
extern "C" void kernel_launch(void* const* d_in, const int* in_sizes, int n_in,
                              void* d_out, int out_size, void* d_ws, size_t ws_size,
                              hipStream_t stream) {
    const float* loc_data  = (const float*)d_in[0];
    const float* conf_data = (const float*)d_in[1];
    const float* priors    = (const float*)d_in[2];
    const float* targets   = (const float*)d_in[3];

    char* ws = (char*)d_ws;
    size_t off = 0;
    float* mined       = (float*)(ws + off); off += (size_t)NB * NP * sizeof(float);
    float* part_loss_l = (float*)(ws + off); off += (size_t)NB * SPLIT * sizeof(float);
    float* part_ce_pos = (float*)(ws + off); off += (size_t)NB * SPLIT * sizeof(float);
    int*   part_npos   = (int*)  (ws + off); off += (size_t)NB * SPLIT * sizeof(int);
    float* row_loss_l  = (float*)(ws + off); off += NB * sizeof(float);
    float* row_ce_pos  = (float*)(ws + off); off += NB * sizeof(float);
    float* row_topk    = (float*)(ws + off); off += NB * sizeof(float);
    int*   row_npos    = (int*)  (ws + off); off += NB * sizeof(int);
    int*   bpi         = (int*)  (ws + off); off += (size_t)NB * NO * sizeof(int);
    (void)ws_size; (void)in_sizes; (void)n_in; (void)out_size;

    k_best_prior<<<NB, 256, 0, stream>>>(targets, priors, bpi);
    k_match_losses<<<NB * SPLIT, 256, 0, stream>>>(loc_data, conf_data, priors, targets,
                                                   bpi, mined, part_loss_l, part_ce_pos,
                                                   part_npos);
    k_row_reduce<<<1, NB, 0, stream>>>(part_loss_l, part_ce_pos, part_npos,
                                       row_loss_l, row_ce_pos, row_npos);
    k_topk_sum<<<NB, 256, 0, stream>>>(mined, row_npos, row_topk);
    k_finalize<<<1, 32, 0, stream>>>(row_loss_l, row_ce_pos, row_topk, row_npos,
                                     (float*)d_out);
}